// TripletLoss_16088947491301
// MI455X (gfx1250) — compile-verified
//
#include <hip/hip_runtime.h>
#include <math.h>

typedef float v2f __attribute__((ext_vector_type(2)));
typedef float v8f __attribute__((ext_vector_type(8)));

#define N_PTS 8192
#define DIM 256
#define MARGIN_F 0.3f
#define NWAVES 4
#define MTILES 2                      // 2 x 16 = 32 anchors per block
#define ANCH (16 * MTILES)
#define FLT_BIG 3.402823466e+38f

// ---------------- Kernel 1: row squared norms (wave per row) ----------------
__global__ __launch_bounds__(256) void tl_row_norms(const float* __restrict__ X,
                                                    float* __restrict__ sq) {
    const int lane = threadIdx.x & 31;
    const int wv   = threadIdx.x >> 5;
    const int row  = blockIdx.x * 8 + wv;
    const float4* p = (const float4*)(X + (size_t)row * DIM);
    float s = 0.f;
#pragma unroll
    for (int t = 0; t < 2; ++t) {          // 64 float4 per row / 32 lanes
        float4 v = p[lane + 32 * t];
        s += v.x * v.x + v.y * v.y + v.z * v.z + v.w * v.w;
    }
#pragma unroll
    for (int m = 16; m > 0; m >>= 1) s += __shfl_xor(s, m, 32);
    if (lane == 0) sq[row] = s;
}

// ---- Kernel 2: 32-anchor block; f32 WMMA Gram + hardest pos/neg (on d^2) ----
__global__ __launch_bounds__(NWAVES * 32) void tl_tile(
        const float* __restrict__ X, const int* __restrict__ lab,
        const float* __restrict__ sq, float* __restrict__ lossOut,
        float* __restrict__ validOut) {
    __shared__ float As[ANCH][260];        // pitch 260: bank-conflict-free
    __shared__ int   labA[ANCH];
    __shared__ float sqA[ANCH];
    __shared__ float redMax[NWAVES][ANCH];
    __shared__ float redMin[NWAVES][ANCH];
    __shared__ int   redCnt[NWAVES][ANCH];

    const int tid = threadIdx.x;
    const int i0  = blockIdx.x * ANCH;

    // Stage anchor tile (32 rows x 256 f32) into LDS, float4 vectorized.
    for (int f = tid; f < ANCH * (DIM / 4); f += NWAVES * 32) {
        int row = f >> 6;                  // 64 float4 per row
        int c4  = f & 63;
        float4 v = ((const float4*)(X + (size_t)(i0 + row) * DIM))[c4];
        *(float4*)&As[row][c4 * 4] = v;
    }
    if (tid < ANCH) { labA[tid] = lab[i0 + tid]; sqA[tid] = sq[i0 + tid]; }
    __syncthreads();

    const int lane = tid & 31;
    const int wv   = tid >> 5;
    const int n    = lane & 15;            // column-within-tile this lane holds
    const int hi   = lane >> 4;            // half-wave select

    int   labI[MTILES][8];
    float sqI[MTILES][8];
#pragma unroll
    for (int t = 0; t < MTILES; ++t)
#pragma unroll
        for (int r = 0; r < 8; ++r) {      // C/D layout: VGPR r -> M = r + 8*hi
            labI[t][r] = labA[16 * t + r + 8 * hi];
            sqI[t][r]  = sqA[16 * t + r + 8 * hi];
        }

    // Track max/min of SQUARED distance (sqrt is monotone; defer to the end).
    float mp[MTILES][8], mn[MTILES][8];
    int   cnt[MTILES][8];
#pragma unroll
    for (int t = 0; t < MTILES; ++t)
#pragma unroll
        for (int r = 0; r < 8; ++r) { mp[t][r] = 0.f; mn[t][r] = FLT_BIG; cnt[t][r] = 0; }

    // Each wave owns tiles j0 = wv*16, wv*16+64, ...  (512/4 = 128 tiles)
    for (int j0 = wv * 16; j0 < N_PTS; j0 += NWAVES * 16) {
        const int   jn   = j0 + n;
        const int   labJ = lab[jn];
        const float sqJ  = sq[jn];
        const float* Brow = X + (size_t)jn * DIM + 2 * hi;

        // prefetch this lane's row for the next tile (clamped, no divergence)
        int pj = jn + NWAVES * 16;
        pj = (pj < N_PTS) ? pj : jn;
        __builtin_prefetch(X + (size_t)pj * DIM, 0, 0);

        v8f acc0 = {};
        v8f acc1 = {};
#pragma unroll 8
        for (int k = 0; k < DIM; k += 4) {
            // B frag (4x16 f32): lane -> col n, K offset 2*hi (+0,+1)
            v2f b  = *(const v2f*)(Brow + k);
            // A frags (16x4 f32): lane -> row n, K offset 2*hi (+0,+1)
            v2f a0 = *(const v2f*)&As[n][k + 2 * hi];
            v2f a1 = *(const v2f*)&As[16 + n][k + 2 * hi];
            acc0 = __builtin_amdgcn_wmma_f32_16x16x4_f32(
                    false, a0, false, b, (short)0, acc0, false, false);
            acc1 = __builtin_amdgcn_wmma_f32_16x16x4_f32(
                    false, a1, false, b, (short)0, acc1, false, false);
        }

        // Branchless epilogue on squared distances (no EXEC divergence).
#pragma unroll
        for (int r = 0; r < 8; ++r) {
            {
                float d2 = sqI[0][r] + sqJ - 2.f * acc0[r];
                d2 = fmaxf(d2, 0.f);
                bool same = (labI[0][r] == labJ);
                cnt[0][r] += same ? 1 : 0;
                mp[0][r] = fmaxf(mp[0][r], same ? d2 : 0.f);
                mn[0][r] = fminf(mn[0][r], same ? FLT_BIG : d2);
            }
            {
                float d2 = sqI[1][r] + sqJ - 2.f * acc1[r];
                d2 = fmaxf(d2, 0.f);
                bool same = (labI[1][r] == labJ);
                cnt[1][r] += same ? 1 : 0;
                mp[1][r] = fmaxf(mp[1][r], same ? d2 : 0.f);
                mn[1][r] = fminf(mn[1][r], same ? FLT_BIG : d2);
            }
        }
    }

    // Reduce across the 16 lanes of each half-wave (N dimension).
#pragma unroll
    for (int s = 1; s < 16; s <<= 1) {
#pragma unroll
        for (int t = 0; t < MTILES; ++t)
#pragma unroll
            for (int r = 0; r < 8; ++r) {
                mp[t][r]  = fmaxf(mp[t][r], __shfl_xor(mp[t][r], s, 32));
                mn[t][r]  = fminf(mn[t][r], __shfl_xor(mn[t][r], s, 32));
                cnt[t][r] += __shfl_xor(cnt[t][r], s, 32);
            }
    }
    if (n == 0) {
#pragma unroll
        for (int t = 0; t < MTILES; ++t)
#pragma unroll
            for (int r = 0; r < 8; ++r) {
                redMax[wv][16 * t + r + 8 * hi] = mp[t][r];
                redMin[wv][16 * t + r + 8 * hi] = mn[t][r];
                redCnt[wv][16 * t + r + 8 * hi] = cnt[t][r];
            }
    }
    __syncthreads();

    // Combine the 4 waves; 2 sqrts per anchor (instead of per pair).
    if (tid < ANCH) {
        float M = 0.f, m = FLT_BIG;
        int   c = 0;
#pragma unroll
        for (int w = 0; w < NWAVES; ++w) {
            M = fmaxf(M, redMax[w][tid]);
            m = fminf(m, redMin[w][tid]);
            c += redCnt[w][tid];
        }
        bool valid = (c > 1) && (m < FLT_BIG) && (M > 0.f);
        float l = valid ? fmaxf(0.f, sqrtf(M) - sqrtf(m) + MARGIN_F) : 0.f;
        lossOut[i0 + tid]  = l;
        validOut[i0 + tid] = valid ? 1.f : 0.f;
    }
}

// ---------------- Kernel 3: final mean over valid anchors -------------------
__global__ __launch_bounds__(256) void tl_finalize(const float* __restrict__ lossBuf,
                                                   const float* __restrict__ validBuf,
                                                   float* __restrict__ out) {
    __shared__ float ssum[256];
    __shared__ float scnt[256];
    float s = 0.f, c = 0.f;
    for (int i = threadIdx.x; i < N_PTS; i += 256) {
        s += lossBuf[i];
        c += validBuf[i];
    }
    ssum[threadIdx.x] = s;
    scnt[threadIdx.x] = c;
    __syncthreads();
    for (int st = 128; st > 0; st >>= 1) {
        if (threadIdx.x < st) {
            ssum[threadIdx.x] += ssum[threadIdx.x + st];
            scnt[threadIdx.x] += scnt[threadIdx.x + st];
        }
        __syncthreads();
    }
    if (threadIdx.x == 0)
        out[0] = (scnt[0] > 0.f) ? (ssum[0] / fmaxf(scnt[0], 1.f)) : 0.f;
}

extern "C" void kernel_launch(void* const* d_in, const int* in_sizes, int n_in,
                              void* d_out, int out_size, void* d_ws, size_t ws_size,
                              hipStream_t stream) {
    const float* X      = (const float*)d_in[0];
    const int*   labels = (const int*)d_in[1];

    float* ws       = (float*)d_ws;
    float* sq       = ws;                 // [8192]
    float* lossBuf  = ws + N_PTS;         // [8192]
    float* validBuf = ws + 2 * N_PTS;     // [8192]

    tl_row_norms<<<N_PTS / 8, 256, 0, stream>>>(X, sq);
    tl_tile<<<N_PTS / ANCH, NWAVES * 32, 0, stream>>>(X, labels, sq, lossBuf, validBuf);
    tl_finalize<<<1, 256, 0, stream>>>(lossBuf, validBuf, (float*)d_out);
}